// VGTCoreV03_75969381532267
// MI455X (gfx1250) — compile-verified
//
#include <hip/hip_runtime.h>
#include <hip/hip_bf16.h>

typedef float v2f __attribute__((ext_vector_type(2)));
typedef float v4f __attribute__((ext_vector_type(4)));
typedef float v8f __attribute__((ext_vector_type(8)));

#define NN_ ((size_t)2048 * 2048)
static constexpr int Nn = 2048;

__device__ __forceinline__ v8f wmma_f32_16x16x4(v2f a, v2f b, v8f c) {
    return __builtin_amdgcn_wmma_f32_16x16x4_f32(false, a, false, b, (short)0, c, false, false);
}

// Cross-lane xor exchange via v_permlane16 / v_permlanex16 (VALU, no LDS pipe).
template <int M>
__device__ __forceinline__ float lane_xor(float x) {
    int v = __float_as_int(x);
    int r;
    if constexpr (M == 16) {        // swap lane halves (identity within half)
        r = __builtin_amdgcn_permlanex16(v, v, 0x76543210, 0xfedcba98, true, false);
    } else if constexpr (M == 1) {  // i ^ 1 within half
        r = __builtin_amdgcn_permlane16(v, v, 0x67452301, 0xefcdab89, true, false);
    } else if constexpr (M == 2) {  // i ^ 2
        r = __builtin_amdgcn_permlane16(v, v, 0x54761032, 0xdcfe98ba, true, false);
    } else if constexpr (M == 4) {  // i ^ 4
        r = __builtin_amdgcn_permlane16(v, v, 0x32107654, 0xba98fedc, true, false);
    } else {                        // i ^ 8
        r = __builtin_amdgcn_permlane16(v, v, 0xfedcba98, 0x76543210, true, false);
    }
    return __int_as_float(r);
}

__device__ __forceinline__ float wave_max(float x) {
    x = fmaxf(x, lane_xor<1>(x));  x = fmaxf(x, lane_xor<2>(x));
    x = fmaxf(x, lane_xor<4>(x));  x = fmaxf(x, lane_xor<8>(x));
    return fmaxf(x, lane_xor<16>(x));
}
__device__ __forceinline__ float wave_sum(float x) {
    x += lane_xor<1>(x);  x += lane_xor<2>(x);
    x += lane_xor<4>(x);  x += lane_xor<8>(x);
    return x + lane_xor<16>(x);
}

// ---------------- Kernel A: QKV projection (X[2048x16] @ W[16x16]) -------------
__global__ __launch_bounds__(256) void qkv_kernel(
    const float* __restrict__ X,
    const float* __restrict__ Wq, const float* __restrict__ bq,
    const float* __restrict__ Wk, const float* __restrict__ bk,
    const float* __restrict__ Wv, const float* __restrict__ bv,
    float* __restrict__ Qw, float* __restrict__ Kw, float* __restrict__ Vw) {
    int idx = blockIdx.x * blockDim.x + threadIdx.x;   // n*16 + c
    if (idx >= Nn * 16) return;
    int n = idx >> 4, c = idx & 15;
    const float* x = X + n * 16;
    float q = bq[c], k = bk[c], v = bv[c];
#pragma unroll
    for (int e = 0; e < 16; e++) {
        float xe = x[e];
        q = fmaf(xe, Wq[e * 16 + c], q);
        k = fmaf(xe, Wk[e * 16 + c], k);
        v = fmaf(xe, Wv[e * 16 + c], v);
    }
    Qw[idx] = q * 0.5f;   // fold 1/sqrt(D) into Q
    Kw[idx] = k;
    Vw[idx] = v;
}

// ---------------- Kernel B: fused QK^T + edge-MLP + mask -> raw scores ----------
// grid = (128 row-tiles, 4 col-groups), block = 256 (8 waves).
// Edge-MLP layer1:  C[hid(16)][pair(16)] = We1^T (16x4) x Edge^T (4x16) via WMMA
//   -> pairs in lanes, hidden in C-register index; layer2 = 8 FMAs per register.
// Raw scores stay RT (L2-resident for the softmax kernel); edge/adj streams NT.
__global__ __launch_bounds__(256) void scores_kernel(
    const float* __restrict__ edge, const float* __restrict__ adj,
    const float* __restrict__ We1, const float* __restrict__ be1,
    const float* __restrict__ We2, const float* __restrict__ be2,
    const float* __restrict__ Qw, const float* __restrict__ Kw,
    float* __restrict__ raw) {
    const int n0   = blockIdx.x * 16;
    const int wave = threadIdx.x >> 5;
    const int lane = threadIdx.x & 31;
    const int lm   = lane & 15;   // column j within tile / matrix N index
    const int lh   = lane >> 4;   // lane half: WMMA K-half + head-within-pair select

    // A operand for edge-MLP layer1: A[M=k'][K=e] = We1[h'][e][unit], k'=8h'+unit
    v2f a_w1a, a_w1b;
    {
        int hp = lm >> 3, un = lm & 7;
        a_w1a.x = We1[((0 + hp) * 4 + (2 * lh + 0)) * 8 + un];
        a_w1a.y = We1[((0 + hp) * 4 + (2 * lh + 1)) * 8 + un];
        a_w1b.x = We1[((2 + hp) * 4 + (2 * lh + 0)) * 8 + un];
        a_w1b.y = We1[((2 + hp) * 4 + (2 * lh + 1)) * 8 + un];
    }
    // C-operand bias (be1) and layer-2 weights; lane-half picks head 0/1 (or 2/3)
    v8f cbias_a, cbias_b;
    float w2ab[8], w2cd[8];
#pragma unroll
    for (int r = 0; r < 8; r++) {
        cbias_a[r] = be1[(0 + lh) * 8 + r];
        cbias_b[r] = be1[(2 + lh) * 8 + r];
        w2ab[r]    = We2[(0 + lh) * 8 + r];
        w2cd[r]    = We2[(2 + lh) * 8 + r];
    }
    // fold +1e10 mask offset into the layer-2 bias
    const float be01p = be2[0 + lh] + 1e10f;
    const float be23p = be2[2 + lh] + 1e10f;

    // Q as WMMA A operand per head (row tile fixed for whole block)
    v2f a_q[4];
#pragma unroll
    for (int h = 0; h < 4; h++) {
        const float* qp = Qw + (size_t)(n0 + lm) * 16 + h * 4 + 2 * lh;
        a_q[h].x = qp[0]; a_q[h].y = qp[1];
    }

    const v8f zero = {0.f, 0.f, 0.f, 0.f, 0.f, 0.f, 0.f, 0.f};

    for (int t = 0; t < 4; t++) {
        const int ctile = blockIdx.y * 32 + t * 8 + wave;   // 0..127
        const int m0 = ctile * 16;

        // QK^T tile per head: B[K=k][N=j] = K[m0+j][h*4+k]
        v8f cqk[4];
#pragma unroll
        for (int h = 0; h < 4; h++) {
            const float* kp = Kw + (size_t)(m0 + lm) * 16 + h * 4 + 2 * lh;
            v2f bk2; bk2.x = kp[0]; bk2.y = kp[1];
            cqk[h] = wmma_f32_16x16x4(a_q[h], bk2, zero);
        }

#pragma unroll
        for (int i = 0; i < 16; i++) {
            // B operand: Edge^T (4x16) for row n0+i: B[e][j] = edge[n0+i][m0+j][e]
            const float* ep = edge + ((size_t)(n0 + i) * Nn + (m0 + lm)) * 4 + 2 * lh;
            v2f be_ = __builtin_nontemporal_load((const v2f*)ep);
            // prefetch this wave's next col-tile (this row advances +8 tiles)
            __builtin_prefetch(ep + 8 * 16 * 4, 0, 1);

            v8f c1a = wmma_f32_16x16x4(a_w1a, be_, cbias_a);   // heads 0/1 hidden
            v8f c1b = wmma_f32_16x16x4(a_w1b, be_, cbias_b);   // heads 2/3 hidden

            float acc01 = be01p, acc23 = be23p;                // edge-MLP layer 2
#pragma unroll
            for (int r = 0; r < 8; r++) {
                acc01 = fmaf(fmaxf(c1a[r], 0.f), w2ab[r], acc01);
                acc23 = fmaf(fmaxf(c1b[r], 0.f), w2cd[r], acc23);
            }

            // extract row i of QK tiles into pair-in-lane layout (heads per half)
            const int r8 = i & 7;
            float q01, q23;
            if (i < 8) {
                float s1 = lane_xor<16>(cqk[1][r8]);
                float s3 = lane_xor<16>(cqk[3][r8]);
                q01 = (lane < 16) ? cqk[0][r8] : s1;
                q23 = (lane < 16) ? cqk[2][r8] : s3;
            } else {
                float s0 = lane_xor<16>(cqk[0][r8]);
                float s2 = lane_xor<16>(cqk[2][r8]);
                q01 = (lane < 16) ? s0 : cqk[1][r8];
                q23 = (lane < 16) ? s2 : cqk[3][r8];
            }

            float adjv = __builtin_nontemporal_load(
                adj + (size_t)(n0 + i) * Nn + m0 + lm);
            // masked = adj*(qk + edge + 1e10) - 1e10  (1e10 folded into acc)
            float m01 = fmaf(adjv, q01 + acc01, -1e10f);
            float m23 = fmaf(adjv, q23 + acc23, -1e10f);

            size_t off = (size_t)lh * NN_ + (size_t)(n0 + i) * Nn + m0 + lm;
            raw[off]           = m01;   // heads 0 / 1 by lane half (L2-resident)
            raw[off + 2 * NN_] = m23;   // heads 2 / 3
        }
    }
}

// ---------------- Kernel C: one wave per (head,row): softmax + attn@V -----------
// Whole 8KB row lives in 64 VGPRs (16 x v4f); V head-slice staged in LDS (32KB).
__global__ __launch_bounds__(256) void softmax_av_kernel(
    const float* __restrict__ Vw, float* __restrict__ attn, float* __restrict__ outTmp) {
    int gw   = (blockIdx.x * blockDim.x + threadIdx.x) >> 5;   // 0..8191
    int lane = threadIdx.x & 31;
    int h = gw >> 11;           // uniform within block (8 waves, 256 blocks/head)
    int n = gw & 2047;

    __shared__ v4f vs[2048];    // V[:, h*4 .. h*4+3]
    for (int m = threadIdx.x; m < 2048; m += 256)
        vs[m] = *(const v4f*)(Vw + (size_t)m * 16 + h * 4);
    __syncthreads();

    float* row = attn + (size_t)h * NN_ + (size_t)n * Nn;
    const v4f* rp = (const v4f*)row;

    v4f x[16];
    float mx = -3.0e38f;
#pragma unroll
    for (int k = 0; k < 16; k++) {
        x[k] = rp[lane + 32 * k];
        mx = fmaxf(mx, fmaxf(fmaxf(x[k].x, x[k].y), fmaxf(x[k].z, x[k].w)));
    }
    mx = wave_max(mx);

    float sum = 0.f;
#pragma unroll
    for (int k = 0; k < 16; k++) {
        x[k].x = __expf(x[k].x - mx); x[k].y = __expf(x[k].y - mx);
        x[k].z = __expf(x[k].z - mx); x[k].w = __expf(x[k].w - mx);
        sum += (x[k].x + x[k].y) + (x[k].z + x[k].w);
    }
    sum = wave_sum(sum);
    float inv = 1.0f / sum;

    float o0 = 0.f, o1 = 0.f, o2 = 0.f, o3 = 0.f;
#pragma unroll
    for (int k = 0; k < 16; k++) {
        v4f a = x[k] * inv;
        __builtin_nontemporal_store(a, (v4f*)rp + lane + 32 * k);  // final attn
        int m = 4 * lane + 128 * k;
        v4f va = vs[m + 0], vb = vs[m + 1], vc = vs[m + 2], vd = vs[m + 3];
        o0 = fmaf(a.x, va.x, fmaf(a.y, vb.x, fmaf(a.z, vc.x, fmaf(a.w, vd.x, o0))));
        o1 = fmaf(a.x, va.y, fmaf(a.y, vb.y, fmaf(a.z, vc.y, fmaf(a.w, vd.y, o1))));
        o2 = fmaf(a.x, va.z, fmaf(a.y, vb.z, fmaf(a.z, vc.z, fmaf(a.w, vd.z, o2))));
        o3 = fmaf(a.x, va.w, fmaf(a.y, vb.w, fmaf(a.z, vc.w, fmaf(a.w, vd.w, o3))));
    }
    o0 = wave_sum(o0); o1 = wave_sum(o1); o2 = wave_sum(o2); o3 = wave_sum(o3);
    if (lane == 0) {
        float* op = outTmp + (size_t)n * 16 + h * 4;
        op[0] = o0; op[1] = o1; op[2] = o2; op[3] = o3;
    }
}

// ---------------- Kernel D: out @ Wo + bo, LayerNorm ---------------------------
__global__ __launch_bounds__(256) void proj_ln_kernel(
    const float* __restrict__ outTmp, const float* __restrict__ Wo,
    const float* __restrict__ bo, const float* __restrict__ g,
    const float* __restrict__ b, float* __restrict__ out) {
    int n = blockIdx.x * blockDim.x + threadIdx.x;
    if (n >= Nn) return;
    float xi[16];
#pragma unroll
    for (int e = 0; e < 16; e++) xi[e] = outTmp[(size_t)n * 16 + e];
    float y[16]; float mu = 0.f;
#pragma unroll
    for (int c = 0; c < 16; c++) {
        float s = bo[c];
#pragma unroll
        for (int e = 0; e < 16; e++) s = fmaf(xi[e], Wo[e * 16 + c], s);
        y[c] = s; mu += s;
    }
    mu *= (1.0f / 16.0f);
    float var = 0.f;
#pragma unroll
    for (int c = 0; c < 16; c++) { float d = y[c] - mu; var = fmaf(d, d, var); }
    var *= (1.0f / 16.0f);
    float rs = rsqrtf(var + 1e-5f);
#pragma unroll
    for (int c = 0; c < 16; c++)
        out[(size_t)n * 16 + c] = (y[c] - mu) * rs * g[c] + b[c];
}

// ---------------- Host-side launcher -------------------------------------------
extern "C" void kernel_launch(void* const* d_in, const int* in_sizes, int n_in,
                              void* d_out, int out_size, void* d_ws, size_t ws_size,
                              hipStream_t stream) {
    const float* X    = (const float*)d_in[0];
    const float* edge = (const float*)d_in[1];
    const float* adj  = (const float*)d_in[2];
    const float* Wq   = (const float*)d_in[3];
    const float* bq   = (const float*)d_in[4];
    const float* Wk   = (const float*)d_in[5];
    const float* bk   = (const float*)d_in[6];
    const float* Wv   = (const float*)d_in[7];
    const float* bv   = (const float*)d_in[8];
    const float* We1  = (const float*)d_in[9];
    const float* be1  = (const float*)d_in[10];
    const float* We2  = (const float*)d_in[11];
    const float* be2  = (const float*)d_in[12];
    const float* Wo   = (const float*)d_in[13];
    const float* bo   = (const float*)d_in[14];
    const float* ln_g = (const float*)d_in[15];
    const float* ln_b = (const float*)d_in[16];

    float* out  = (float*)d_out;              // [2048,16]
    float* attn = out + (size_t)Nn * 16;      // [4,2048,2048] (raw scores -> attn)

    float* ws   = (float*)d_ws;
    float* Qw   = ws;                         // [2048,16]
    float* Kw   = ws + 32768;                 // [2048,16]
    float* Vw   = ws + 65536;                 // [2048,16]
    float* oT   = ws + 98304;                 // [2048,16]

    qkv_kernel<<<128, 256, 0, stream>>>(X, Wq, bq, Wk, bk, Wv, bv, Qw, Kw, Vw);
    scores_kernel<<<dim3(128, 4), 256, 0, stream>>>(edge, adj, We1, be1, We2, be2,
                                                    Qw, Kw, attn);
    softmax_av_kernel<<<1024, 256, 0, stream>>>(Vw, attn, oT);
    proj_ln_kernel<<<8, 256, 0, stream>>>(oT, Wo, bo, ln_g, ln_b, out);
}